// GhostGradientEstimator_90726889161024
// MI455X (gfx1250) — compile-verified
//
#include <hip/hip_runtime.h>

// LSS gradient via block-tridiagonal KKT solve, parallel cyclic reduction (PCR).
// All 8x8-block products batched in pairs onto V_WMMA_F32_16X16X4_F32 (f32 WMMA).

typedef __attribute__((ext_vector_type(2))) float v2f;
typedef __attribute__((ext_vector_type(8))) float v8f;

#define N_T   1024
#define HALO  512
#define NPAD  2048
#define EPSR  1e-6f

// ---------- WMMA helper: 16x16x16 f32 matmul from LDS tiles (row-major 16x16) ----------
__device__ __forceinline__ v8f wmma16(const float* As, const float* Bs, v8f acc) {
  const int lane = threadIdx.x & 31;
  const int hi   = (lane >> 4) & 1;
  const int l    = lane & 15;
#pragma unroll
  for (int k0 = 0; k0 < 16; k0 += 4) {
    const int ka = k0 + 2 * hi;          // A: lanes 0-15 -> K=k0,k0+1 ; 16-31 -> K=k0+2,k0+3
    v2f a, b;
    a.x = As[l * 16 + ka];
    a.y = As[l * 16 + ka + 1];
    b.x = Bs[ka * 16 + l];               // B: row K, col l
    b.y = Bs[(ka + 1) * 16 + l];
    acc = __builtin_amdgcn_wmma_f32_16x16x4_f32(false, a, false, b, (short)0,
                                                acc, false, false);
  }
  return acc;
}

__device__ __forceinline__ v8f zero8() { v8f z = {}; return z; }

// stage blockdiag(scale*B0, scale*B1) of two 8x8 row-major blocks into 16x16 LDS tile
__device__ __forceinline__ void stage_bd(float* T, const float* B0, const float* B1,
                                         float scale, int lane) {
#pragma unroll
  for (int idx = lane; idx < 256; idx += 32) {
    const int r = idx >> 4, c = idx & 15;
    float v = 0.f;
    if (r < 8 && c < 8)        v = scale * B0[r * 8 + c];
    else if (r >= 8 && c >= 8) v = scale * B1[(r - 8) * 8 + (c - 8)];
    T[idx] = v;
  }
}

// stage stacked [R0 ; R1] of two 8x16 row-major blocks into 16x16 LDS tile
__device__ __forceinline__ void stage_st(float* T, const float* R0, const float* R1,
                                         int lane) {
#pragma unroll
  for (int idx = lane; idx < 256; idx += 32) {
    const int r = idx >> 4, c = idx & 15;
    T[idx] = (r < 8) ? R0[r * 16 + c] : R1[(r - 8) * 16 + c];
  }
}

// accumulator <-> memory (acc element: row = v + 8*hi, col = lane&15)
__device__ __forceinline__ v8f loadC_bd(const float* B0, const float* B1, int lane) {
  const int hi = (lane >> 4) & 1, col = lane & 15;
  v8f c;
#pragma unroll
  for (int v = 0; v < 8; ++v) {
    float x = 0.f;
    if (!hi && col < 8)       x = B0[v * 8 + col];
    else if (hi && col >= 8)  x = B1[v * 8 + (col - 8)];
    c[v] = x;
  }
  return c;
}
__device__ __forceinline__ void storeD_bd(float* B0, float* B1, v8f acc, int lane) {
  const int hi = (lane >> 4) & 1, col = lane & 15;
#pragma unroll
  for (int v = 0; v < 8; ++v) {
    if (!hi && col < 8)       B0[v * 8 + col] = acc[v];
    else if (hi && col >= 8)  B1[v * 8 + (col - 8)] = acc[v];
  }
}
__device__ __forceinline__ v8f loadC_st(const float* R0, const float* R1, int lane) {
  const int hi = (lane >> 4) & 1, col = lane & 15;
  v8f c;
#pragma unroll
  for (int v = 0; v < 8; ++v) c[v] = hi ? R1[v * 16 + col] : R0[v * 16 + col];
  return c;
}
__device__ __forceinline__ void store_st(float* R0, float* R1, v8f acc, int lane) {
  const int hi = (lane >> 4) & 1, col = lane & 15;
#pragma unroll
  for (int v = 0; v < 8; ++v) {
    if (hi) R1[v * 16 + col] = acc[v];
    else    R0[v * 16 + col] = acc[v];
  }
}
// dump accumulator into row-major 16x16 LDS tile (for reuse as A operand)
__device__ __forceinline__ void storeT(float* T, v8f acc, int lane) {
  const int hi = (lane >> 4) & 1, col = lane & 15;
#pragma unroll
  for (int v = 0; v < 8; ++v) T[(v + 8 * hi) * 16 + col] = acc[v];
}

// ---------- kernels ----------
__global__ void k_zero(float* p, long n) {
  long i = (long)blockIdx.x * blockDim.x + threadIdx.x;
  long stride = (long)gridDim.x * blockDim.x;
  for (; i < n; i += stride) p[i] = 0.f;
}

// Build D = (1+eps)I + M M^T (via WMMA), L = -M_t, U = -M_{t+1}^T, r = -P_t
__global__ void __launch_bounds__(256) k_build(const float* __restrict__ M,
                                               const float* __restrict__ P,
                                               float* D, float* L, float* U, float* R) {
  __shared__ float lds[8][2][256];
  const int lane = threadIdx.x & 31, w = threadIdx.x >> 5;
  const int j = blockIdx.x * 8 + w;           // pair id 0..511
  const int t0 = 2 * j, t1 = t0 + 1;
  float* TA = lds[w][0];
  float* TB = lds[w][1];
  const float* M0 = M + (size_t)t0 * 64;
  const float* M1 = M + (size_t)t1 * 64;

  stage_bd(TA, M0, M1, 1.f, lane);            // blockdiag(M0, M1)
#pragma unroll
  for (int idx = lane; idx < 256; idx += 32) { // blockdiag(M0^T, M1^T)
    const int r = idx >> 4, c = idx & 15;
    float v = 0.f;
    if (r < 8 && c < 8)        v = M0[c * 8 + r];
    else if (r >= 8 && c >= 8) v = M1[(c - 8) * 8 + (r - 8)];
    TB[idx] = v;
  }
  __syncthreads();

  const int hi = (lane >> 4) & 1, col = lane & 15;
  v8f c;
#pragma unroll
  for (int v = 0; v < 8; ++v) c[v] = ((v + 8 * hi) == col) ? (1.0f + EPSR) : 0.f;
  c = wmma16(TA, TB, c);
  storeD_bd(D + (size_t)(HALO + t0) * 64, D + (size_t)(HALO + t1) * 64, c, lane);

#pragma unroll
  for (int tt = 0; tt < 2; ++tt) {
    const int t = tt ? t1 : t0;
    const size_t q = (size_t)(HALO + t);
    for (int e = lane; e < 64; e += 32) {
      const int i = e >> 3, jj = e & 7;
      L[q * 64 + e] = (t >= 1) ? -M[(size_t)t * 64 + e] : 0.f;
      U[q * 64 + e] = (t <= N_T - 2) ? -M[(size_t)(t + 1) * 64 + jj * 8 + i] : 0.f;
    }
    for (int e = lane; e < 128; e += 32) R[q * 128 + e] = -P[(size_t)t * 128 + e];
  }
}

// Per-thread in-place Gauss-Jordan inverse of SPD 8x8 blocks (registers only)
__global__ void __launch_bounds__(256) k_inv(const float* __restrict__ D, float* DI) {
  const int t = blockIdx.x * blockDim.x + threadIdx.x;
  if (t >= N_T) return;
  const float* src = D + (size_t)(HALO + t) * 64;
  float a[8][8];
#pragma unroll
  for (int i = 0; i < 8; ++i)
#pragma unroll
    for (int jj = 0; jj < 8; ++jj) a[i][jj] = src[i * 8 + jj];
#pragma unroll
  for (int k = 0; k < 8; ++k) {
    const float p = 1.0f / a[k][k];
#pragma unroll
    for (int i = 0; i < 8; ++i) if (i != k) a[i][k] = -a[i][k] * p;
#pragma unroll
    for (int i = 0; i < 8; ++i) if (i != k)
#pragma unroll
      for (int jj = 0; jj < 8; ++jj) if (jj != k) a[i][jj] += a[i][k] * a[k][jj];
#pragma unroll
    for (int jj = 0; jj < 8; ++jj) if (jj != k) a[k][jj] *= p;
    a[k][k] = p;
  }
  float* dst = DI + (size_t)(HALO + t) * 64;
#pragma unroll
  for (int i = 0; i < 8; ++i)
#pragma unroll
    for (int jj = 0; jj < 8; ++jj) dst[i * 8 + jj] = a[i][jj];
}

// One PCR level, stride s. One wave handles rows (2j, 2j+1) batched as 16x16 tiles.
__global__ void __launch_bounds__(256) k_pcr(
    const float* __restrict__ Dc, const float* __restrict__ Lc,
    const float* __restrict__ Uc, const float* __restrict__ Rc,
    float* Dn, float* Ln, float* Un, float* Rn,
    const float* __restrict__ DI, int s) {
  __shared__ float lds[8][4][256];
  const int lane = threadIdx.x & 31, w = threadIdx.x >> 5;
  const int j = blockIdx.x * 8 + w;
  const size_t q0 = (size_t)(HALO + 2 * j), q1 = q0 + 1;
  float* TA  = lds[w][0];
  float* TB  = lds[w][1];
  float* Tal = lds[w][2];
  float* Tbe = lds[w][3];
  v8f acc;

  // alpha = -L * Dinv[-s]
  stage_bd(TA, Lc + q0 * 64, Lc + q1 * 64, -1.f, lane);
  stage_bd(TB, DI + (q0 - s) * 64, DI + (q1 - s) * 64, 1.f, lane);
  __syncthreads();
  acc = wmma16(TA, TB, zero8());
  storeT(Tal, acc, lane);
  __syncthreads();

  // beta = -U * Dinv[+s]
  stage_bd(TA, Uc + q0 * 64, Uc + q1 * 64, -1.f, lane);
  stage_bd(TB, DI + (q0 + s) * 64, DI + (q1 + s) * 64, 1.f, lane);
  __syncthreads();
  acc = wmma16(TA, TB, zero8());
  storeT(Tbe, acc, lane);
  __syncthreads();

  // L' = alpha * L[-s]
  stage_bd(TB, Lc + (q0 - s) * 64, Lc + (q1 - s) * 64, 1.f, lane);
  __syncthreads();
  acc = wmma16(Tal, TB, zero8());
  storeD_bd(Ln + q0 * 64, Ln + q1 * 64, acc, lane);
  __syncthreads();

  // U' = beta * U[+s]
  stage_bd(TB, Uc + (q0 + s) * 64, Uc + (q1 + s) * 64, 1.f, lane);
  __syncthreads();
  acc = wmma16(Tbe, TB, zero8());
  storeD_bd(Un + q0 * 64, Un + q1 * 64, acc, lane);
  __syncthreads();

  // D' = D + alpha*U[-s] + beta*L[+s]
  stage_bd(TB, Uc + (q0 - s) * 64, Uc + (q1 - s) * 64, 1.f, lane);
  __syncthreads();
  acc = loadC_bd(Dc + q0 * 64, Dc + q1 * 64, lane);
  acc = wmma16(Tal, TB, acc);
  __syncthreads();
  stage_bd(TB, Lc + (q0 + s) * 64, Lc + (q1 + s) * 64, 1.f, lane);
  __syncthreads();
  acc = wmma16(Tbe, TB, acc);
  storeD_bd(Dn + q0 * 64, Dn + q1 * 64, acc, lane);
  __syncthreads();

  // r' = r + alpha*r[-s] + beta*r[+s]
  stage_st(TB, Rc + (q0 - s) * 128, Rc + (q1 - s) * 128, lane);
  __syncthreads();
  acc = loadC_st(Rc + q0 * 128, Rc + q1 * 128, lane);
  acc = wmma16(Tal, TB, acc);
  __syncthreads();
  stage_st(TB, Rc + (q0 + s) * 128, Rc + (q1 + s) * 128, lane);
  __syncthreads();
  acc = wmma16(Tbe, TB, acc);
  store_st(Rn + q0 * 128, Rn + q1 * 128, acc, lane);
}

// lambda_t = Dinv_t * r_t  (fully diagonal after PCR), WMMA-batched pairs
__global__ void __launch_bounds__(256) k_solve(const float* __restrict__ DI,
                                               const float* __restrict__ R,
                                               float* LAM) {
  __shared__ float lds[8][2][256];
  const int lane = threadIdx.x & 31, w = threadIdx.x >> 5;
  const int j = blockIdx.x * 8 + w;
  const int t0 = 2 * j, t1 = t0 + 1;
  const size_t q0 = (size_t)(HALO + t0), q1 = q0 + 1;
  float* TA = lds[w][0];
  float* TB = lds[w][1];
  stage_bd(TA, DI + q0 * 64, DI + q1 * 64, 1.f, lane);
  stage_st(TB, R + q0 * 128, R + q1 * 128, lane);
  __syncthreads();
  v8f acc = wmma16(TA, TB, zero8());
  store_st(LAM + (size_t)t0 * 128, LAM + (size_t)t1 * 128, acc, lane);
}

// g_p = (1/1025) * sum_t (M_t dO[t] - dO[t+1])^T lam_t   (single-block reduction)
__global__ void __launch_bounds__(256) k_grad(const float* __restrict__ M,
                                              const float* __restrict__ dO,
                                              const float* __restrict__ LAM,
                                              float* out) {
  __shared__ float red[256 * 16];
  const int tid = threadIdx.x;
  float acc[16];
#pragma unroll
  for (int p = 0; p < 16; ++p) acc[p] = 0.f;
  for (int t = tid; t < N_T; t += 256) {
    float dt[8], c[8];
#pragma unroll
    for (int i = 0; i < 8; ++i) dt[i] = dO[t * 8 + i];
#pragma unroll
    for (int i = 0; i < 8; ++i) {
      float s = 0.f;
#pragma unroll
      for (int jj = 0; jj < 8; ++jj) s += M[(size_t)t * 64 + i * 8 + jj] * dt[jj];
      c[i] = s - dO[(t + 1) * 8 + i];
    }
#pragma unroll
    for (int i = 0; i < 8; ++i) {
      const float ci = c[i];
#pragma unroll
      for (int p = 0; p < 16; ++p) acc[p] += ci * LAM[(size_t)t * 128 + i * 16 + p];
    }
  }
#pragma unroll
  for (int p = 0; p < 16; ++p) red[tid * 16 + p] = acc[p];
  __syncthreads();
  for (int off = 128; off >= 1; off >>= 1) {
    if (tid < off)
#pragma unroll
      for (int p = 0; p < 16; ++p) red[tid * 16 + p] += red[(tid + off) * 16 + p];
    __syncthreads();
  }
  if (tid < 16) out[tid] = red[tid] * (1.0f / 1025.0f);
}

extern "C" void kernel_launch(void* const* d_in, const int* in_sizes, int n_in,
                              void* d_out, int out_size, void* d_ws, size_t ws_size,
                              hipStream_t stream) {
  const float* M  = (const float*)d_in[0];   // (1024, 8, 8)
  const float* P  = (const float*)d_in[1];   // (1024, 8, 16)
  const float* dO = (const float*)d_in[2];   // (1025, 8)
  float* out = (float*)d_out;                // (16,)

  const size_t sz64 = (size_t)NPAD * 64, sz128 = (size_t)NPAD * 128;
  float* p = (float*)d_ws;
  float* Db[2]; float* Lb[2]; float* Ub[2]; float* Rb[2];
  Db[0] = p; p += sz64;  Db[1] = p; p += sz64;
  Lb[0] = p; p += sz64;  Lb[1] = p; p += sz64;
  Ub[0] = p; p += sz64;  Ub[1] = p; p += sz64;
  float* DIv = p; p += sz64;
  Rb[0] = p; p += sz128; Rb[1] = p; p += sz128;
  float* LAM = p; p += (size_t)N_T * 128;
  const long total = (long)(p - (float*)d_ws);

  // zero everything (halos must be 0; also makes replays deterministic)
  k_zero<<<512, 256, 0, stream>>>((float*)d_ws, total);
  k_build<<<64, 256, 0, stream>>>(M, P, Db[0], Lb[0], Ub[0], Rb[0]);

  int cur = 0;
  for (int k = 0; k < 10; ++k) {
    k_inv<<<4, 256, 0, stream>>>(Db[cur], DIv);
    k_pcr<<<64, 256, 0, stream>>>(Db[cur], Lb[cur], Ub[cur], Rb[cur],
                                  Db[1 - cur], Lb[1 - cur], Ub[1 - cur], Rb[1 - cur],
                                  DIv, 1 << k);
    cur = 1 - cur;
  }
  k_inv<<<4, 256, 0, stream>>>(Db[cur], DIv);
  k_solve<<<64, 256, 0, stream>>>(DIv, Rb[cur], LAM);
  k_grad<<<1, 256, 0, stream>>>(M, dO, LAM, out);
}